// GNN_22385369546840
// MI455X (gfx1250) — compile-verified
//
#include <hip/hip_runtime.h>

// GCN 2-layer, N=100000 nodes, E=5000000 edges, H=16.
// Math collapses to: deg scatter -> dinv -> scalar scatter s ->
// per-node 16-wide transform (WMMA f32 16x16x4) -> scalar scatter out.

#define NN 100000
#define NE 5000000
#define NE4 (NE / 4)

typedef __attribute__((ext_vector_type(2))) float v2f;
typedef __attribute__((ext_vector_type(8))) float v8f;

// ---------------- deg: self-loop init + edge-weight scatter ----------------

__global__ void k_init_deg(float* __restrict__ deg) {
  int n = blockIdx.x * blockDim.x + threadIdx.x;
  if (n < NN) deg[n] = 1.0f;  // self-loop weight = 1
}

__global__ void k_deg_scatter(const int* __restrict__ col,
                              const float* __restrict__ w,
                              float* __restrict__ deg) {
  int t = blockIdx.x * blockDim.x + threadIdx.x;
  if (t < NE4) {
    int4   c  = ((const int4*)col)[t];
    float4 wv = ((const float4*)w)[t];
    atomicAdd(&deg[c.x], wv.x);
    atomicAdd(&deg[c.y], wv.y);
    atomicAdd(&deg[c.z], wv.z);
    atomicAdd(&deg[c.w], wv.w);
  }
}

// ---- dinv = rsqrt(deg); seed layer-1 scatter with self-loop term ----------

__global__ void k_dinv_s0(const float* __restrict__ deg,
                          const float* __restrict__ x,
                          float* __restrict__ dinv,
                          float* __restrict__ s) {
  int n = blockIdx.x * blockDim.x + threadIdx.x;
  if (n < NN) {
    float d  = deg[n];
    float di = (d > 0.0f) ? rsqrtf(d) : 0.0f;
    dinv[n] = di;
    s[n] = di * di * x[n];  // self-loop: norm = dinv*1*dinv
  }
}

// ---------------- layer-1 scalar scatter: s[col] += norm * x[row] ----------

__global__ void k_s_scatter(const int* __restrict__ row,
                            const int* __restrict__ col,
                            const float* __restrict__ w,
                            const float* __restrict__ dinv,
                            const float* __restrict__ x,
                            float* __restrict__ s) {
  int t = blockIdx.x * blockDim.x + threadIdx.x;
  if (t < NE4) {
    int4   r  = ((const int4*)row)[t];
    int4   c  = ((const int4*)col)[t];
    float4 wv = ((const float4*)w)[t];
    atomicAdd(&s[c.x], dinv[r.x] * wv.x * dinv[c.x] * x[r.x]);
    atomicAdd(&s[c.y], dinv[r.y] * wv.y * dinv[c.y] * x[r.y]);
    atomicAdd(&s[c.z], dinv[r.z] * wv.z * dinv[c.z] * x[r.z]);
    atomicAdd(&s[c.w], dinv[r.w] * wv.w * dinv[c.w] * x[r.w]);
  }
}

// ---- per-node transform via WMMA f32 16x16x4 (transposed formulation):
//      D[k,m] = W1[k]*s[m] + b1[k]   (features on M axis, nodes on N axis)
//      lane l (node m=l&15, half=l>>4) holds h[m, k=half*8+j] in d[j],
//      so z[m] = sum_k relu(h)*W2[k] is 8 in-lane FMAs + ONE shfl_xor(16)
//      + one coalesced 16-lane store. -------------------------------------

__global__ void __launch_bounds__(320)
k_node_transform(const float* __restrict__ s,
                 const float* __restrict__ W1,
                 const float* __restrict__ b1,
                 const float* __restrict__ W2,
                 float* __restrict__ z) {
  int lane = threadIdx.x & 31;
  int wave = (blockIdx.x * blockDim.x + threadIdx.x) >> 5;  // 6250 waves total
  int base = wave * 16;          // 16-node tile; 100000 = 6250*16 exactly
  int half = lane >> 4;          // 0: lanes 0-15, 1: lanes 16-31
  int l16  = lane & 15;

  // A (16x4 f32): A[k][0] = W1[k]; M=l16 over lanes, VGPR0={K0|K2}, VGPR1={K1|K3}.
  float w1 = W1[l16];
  v2f a;
  a.x = half ? 0.0f : w1;   // only K=0 column nonzero
  a.y = 0.0f;
  // B (4x16 f32): B[0][m] = s[m]; row K=0 at VGPR0, lanes 0-15; rest zero.
  float sv = s[base + l16];
  v2f b;
  b.x = half ? 0.0f : sv;
  b.y = 0.0f;
  // C (16x16 f32): C[k][m] = b1[k]; c[j] = b1[half*8 + j] (same across N lanes).
  const float* b1p = b1 + half * 8;
  v8f c;
#pragma unroll
  for (int j = 0; j < 8; ++j) c[j] = b1p[j];

  // D[k][m] = W1[k]*s[m] + b1[k]; VGPR j holds M=k=half*8+j, N=m=l16.
  v8f d = __builtin_amdgcn_wmma_f32_16x16x4_f32(
      /*neg_a=*/false, a, /*neg_b=*/false, b,
      /*c_mod=*/(short)0, c, /*reuse_a=*/false, /*reuse_b=*/false);

  // z partial: this lane covers k = half*8 .. half*8+7 for node base+l16.
  const float* w2p = W2 + half * 8;
  float t = 0.0f;
#pragma unroll
  for (int j = 0; j < 8; ++j) t = fmaf(fmaxf(d[j], 0.0f), w2p[j], t);

  // combine the two k-halves (lane l <-> l^16), then one coalesced store.
  t += __shfl_xor(t, 16, 32);
  if (half == 0) z[base + l16] = t;
}

// ---- out init: self-loop term of layer 2 + bias (full overwrite of d_out) -

__global__ void k_out_init(const float* __restrict__ dinv,
                           const float* __restrict__ z,
                           const float* __restrict__ b2,
                           float* __restrict__ out) {
  int n = blockIdx.x * blockDim.x + threadIdx.x;
  if (n < NN) {
    float di = dinv[n];
    out[n] = di * di * z[n] + b2[0];
  }
}

// ---------------- layer-2 scalar scatter: out[col] += norm * z[row] --------

__global__ void k_out_scatter(const int* __restrict__ row,
                              const int* __restrict__ col,
                              const float* __restrict__ w,
                              const float* __restrict__ dinv,
                              const float* __restrict__ z,
                              float* __restrict__ out) {
  int t = blockIdx.x * blockDim.x + threadIdx.x;
  if (t < NE4) {
    int4   r  = ((const int4*)row)[t];
    int4   c  = ((const int4*)col)[t];
    float4 wv = ((const float4*)w)[t];
    atomicAdd(&out[c.x], dinv[r.x] * wv.x * dinv[c.x] * z[r.x]);
    atomicAdd(&out[c.y], dinv[r.y] * wv.y * dinv[c.y] * z[r.y]);
    atomicAdd(&out[c.z], dinv[r.z] * wv.z * dinv[c.z] * z[r.z]);
    atomicAdd(&out[c.w], dinv[r.w] * wv.w * dinv[c.w] * z[r.w]);
  }
}

extern "C" void kernel_launch(void* const* d_in, const int* in_sizes, int n_in,
                              void* d_out, int out_size, void* d_ws, size_t ws_size,
                              hipStream_t stream) {
  const float* x   = (const float*)d_in[0];       // [N,1]
  const int*   ei  = (const int*)d_in[1];         // [2,E] flat: row=ei, col=ei+NE
  const float* ew  = (const float*)d_in[2];       // [E]
  const float* W1  = (const float*)d_in[3];       // [1,16]
  const float* b1  = (const float*)d_in[4];       // [16]
  const float* W2  = (const float*)d_in[5];       // [16,1]
  const float* b2  = (const float*)d_in[6];       // [1]
  float*       out = (float*)d_out;               // [N,1]

  const int* row = ei;
  const int* col = ei + NE;

  // Workspace layout (needs 4*N floats = 1.6 MB)
  float* deg  = (float*)d_ws;
  float* dinv = deg + NN;
  float* s    = dinv + NN;
  float* z    = s + NN;

  const int TB = 256;
  const int gridN = (NN + TB - 1) / TB;      // 391
  const int gridE = (NE4 + TB - 1) / TB;     // 4883

  k_init_deg<<<gridN, TB, 0, stream>>>(deg);
  k_deg_scatter<<<gridE, TB, 0, stream>>>(col, ew, deg);
  k_dinv_s0<<<gridN, TB, 0, stream>>>(deg, x, dinv, s);
  k_s_scatter<<<gridE, TB, 0, stream>>>(row, col, ew, dinv, x, s);
  // 6250 waves of 16 nodes; 320 threads = 10 waves/block -> 625 blocks exactly
  k_node_transform<<<625, 320, 0, stream>>>(s, W1, b1, W2, z);
  k_out_init<<<gridN, TB, 0, stream>>>(dinv, z, b2, out);
  k_out_scatter<<<gridE, TB, 0, stream>>>(row, col, ew, dinv, z, out);
}